// LinearSum_11089605558540
// MI455X (gfx1250) — compile-verified
//
#include <hip/hip_runtime.h>
#include <stdint.h>

typedef __attribute__((ext_vector_type(2))) float v2f;
typedef __attribute__((ext_vector_type(4))) float v4f;
typedef __attribute__((ext_vector_type(8))) float v8f;

#define M_TOTAL 32768      // B*N = 64*512
#define TOKS    1024
#define EMB     1024
#define NKP     17
#define APP_F   (EMB + NKP)               // 1041
#define ST_F    (4 + 3 * NKP)             // 55
#define K_TOTAL (APP_F + ST_F)            // 1096
#define K_PAD   1104                      // 69 * 16
#define KTILE   16
#define NTILES  (K_PAD / KTILE)           // 69
#define FAST_NT (EMB / KTILE)             // 64 tiles fully inside emb/app_W
#define BM      128
#define BN      128
#define LDSS    20                        // row stride (floats): 80B -> b128-aligned quads

__device__ __forceinline__ float gatherA(int row, int f,
    const float* __restrict__ emb, const float* __restrict__ vis,
    const float* __restrict__ bbox, const float* __restrict__ kpt) {
  if (f < EMB)         return emb[row * EMB + f];
  if (f < APP_F)       return vis[row * NKP + (f - EMB)];
  if (f < APP_F + 4)   return bbox[row * 4 + (f - APP_F)];
  if (f < K_TOTAL)     return kpt[row * (3 * NKP) + (f - APP_F - 4)];
  return 0.0f;                                  // zero pad K 1096..1103
}

__device__ __forceinline__ float gatherB(int t, int f,
    const float* __restrict__ appW, const float* __restrict__ stW) {
  if (f < APP_F)   return appW[t * APP_F + f];
  if (f < K_TOTAL) return stW[t * ST_F + (f - APP_F)];
  return 0.0f;
}

__global__ __launch_bounds__(256) void fused_token_gemm(
    const float* __restrict__ emb, const float* __restrict__ vis,
    const float* __restrict__ bbox, const float* __restrict__ kpt,
    const uint8_t* __restrict__ mask,
    const float* __restrict__ appW, const float* __restrict__ appB,
    const float* __restrict__ stW,  const float* __restrict__ stB,
    float* __restrict__ out)
{
  __shared__ __align__(16) float As[2][BM][LDSS];
  __shared__ __align__(16) float Bs[2][BN][LDSS];

  const int tid  = threadIdx.x;
  const int lane = tid & 31;
  const int wave = tid >> 5;

  const int wg_m = blockIdx.y * BM;   // row block in M = B*N
  const int wg_n = blockIdx.x * BN;   // col block in TOK

  // 8 waves -> 2 (M) x 4 (N); each wave: 64x32 = 4x2 fragments of 16x16
  const int wm = (wave & 1) * 64;
  const int wn = (wave >> 1) * 32;

  // loader mapping: 128 rows x 4 k-quads = 512 quads / 256 threads = 2 each
  const int q0 = (tid & 3) * 4;       // k offset of this thread's quad
  const int r0 = tid >> 2;            // row for p=0 (0..63); p=1 adds 64

  v4f aQ[2], bQ[2];

  auto load_tile = [&](int kt) {
    const int kb = kt * KTILE;
    if (kt < FAST_NT) {
      // whole tile inside embeddings / app_W: branch-free
      #pragma unroll
      for (int p = 0; p < 2; ++p) {
        const int r = r0 + p * 64;
        aQ[p] = *(const v4f*)&emb[(size_t)(wg_m + r) * EMB + kb + q0]; // 16B aligned
        const float* wr = &appW[(size_t)(wg_n + r) * APP_F + kb + q0]; // unaligned rows
        v4f b; b.x = wr[0]; b.y = wr[1]; b.z = wr[2]; b.w = wr[3];
        bQ[p] = b;
      }
    } else {
      // ragged tail: vis/bbox/kp/st_W + zero pad
      #pragma unroll
      for (int p = 0; p < 2; ++p) {
        const int r = r0 + p * 64;
        v4f a, b;
        #pragma unroll
        for (int j = 0; j < 4; ++j) {
          const int f = kb + q0 + j;
          a[j] = gatherA(wg_m + r, f, emb, vis, bbox, kpt);
          b[j] = gatherB(wg_n + r, f, appW, stW);
        }
        aQ[p] = a; bQ[p] = b;
      }
    }
  };
  auto stage_tile = [&](int buf) {
    #pragma unroll
    for (int p = 0; p < 2; ++p) {
      const int r = r0 + p * 64;
      *(v4f*)&As[buf][r][q0] = aQ[p];   // one ds_store_b128 each
      *(v4f*)&Bs[buf][r][q0] = bQ[p];
    }
  };

  v8f acc[4][2];
  {
    v8f z = {};
    #pragma unroll
    for (int mi = 0; mi < 4; ++mi)
      #pragma unroll
      for (int ni = 0; ni < 2; ++ni)
        acc[mi][ni] = z;
  }

  load_tile(0);
  stage_tile(0);
  __syncthreads();

  const int mrow = lane & 15;          // M (or N) index within fragment
  const int kp2  = (lane >> 4) * 2;    // K pair selected by lane half

  for (int kt = 0; kt < NTILES; ++kt) {
    const int cur = kt & 1;
    if (kt + 1 < NTILES) load_tile(kt + 1);   // global prefetch overlaps WMMA

    #pragma unroll
    for (int s = 0; s < 4; ++s) {            // 4 k-steps of K=4
      const int k0 = s * 4 + kp2;
      v2f afrag[4], bfrag[2];
      #pragma unroll
      for (int mi = 0; mi < 4; ++mi)
        afrag[mi] = *(const v2f*)&As[cur][wm + mi * 16 + mrow][k0];
      #pragma unroll
      for (int ni = 0; ni < 2; ++ni)
        bfrag[ni] = *(const v2f*)&Bs[cur][wn + ni * 16 + mrow][k0];
      #pragma unroll
      for (int mi = 0; mi < 4; ++mi)
        #pragma unroll
        for (int ni = 0; ni < 2; ++ni)
          acc[mi][ni] = __builtin_amdgcn_wmma_f32_16x16x4_f32(
              false, afrag[mi], false, bfrag[ni],
              (short)0, acc[mi][ni], false, false);
    }
    __syncthreads();
    if (kt + 1 < NTILES) {
      stage_tile((kt + 1) & 1);
      __syncthreads();
    }
  }

  // Epilogue: D layout — VGPR i: lanes 0-15 => M=i, lanes 16-31 => M=i+8; N = lane&15
  const int ncol  = lane & 15;
  const int mhalf = (lane >> 4) * 8;
  #pragma unroll
  for (int ni = 0; ni < 2; ++ni) {
    const int t    = wg_n + wn + ni * 16 + ncol;
    const float bb = appB[t] + stB[t];
    #pragma unroll
    for (int mi = 0; mi < 4; ++mi) {
      #pragma unroll
      for (int i = 0; i < 8; ++i) {
        const int gr = wg_m + wm + mi * 16 + mhalf + i;
        const float v = mask[gr] ? (acc[mi][ni][i] + bb) : 0.0f;
        out[(long)gr * TOKS + t] = v;
      }
    }
  }
}

extern "C" void kernel_launch(void* const* d_in, const int* in_sizes, int n_in,
                              void* d_out, int out_size, void* d_ws, size_t ws_size,
                              hipStream_t stream) {
  (void)in_sizes; (void)n_in; (void)out_size; (void)d_ws; (void)ws_size;
  const float*   emb  = (const float*)d_in[0];
  const float*   vis  = (const float*)d_in[1];
  const float*   bbox = (const float*)d_in[2];
  const float*   kpt  = (const float*)d_in[3];
  const uint8_t* mask = (const uint8_t*)d_in[4];   // jax bool -> 1 byte/elem
  const float*   appW = (const float*)d_in[5];
  const float*   appB = (const float*)d_in[6];
  const float*   stW  = (const float*)d_in[7];
  const float*   stB  = (const float*)d_in[8];
  float* out = (float*)d_out;

  dim3 grid(TOKS / BN, M_TOTAL / BM);   // (8, 256)
  fused_token_gemm<<<grid, 256, 0, stream>>>(emb, vis, bbox, kpt, mask,
                                             appW, appB, stW, stB, out);
}